// PlasticEdges_66872640799260
// MI455X (gfx1250) — compile-verified
//
#include <hip/hip_runtime.h>
#include <stdint.h>

#define NN    16
#define CC    4
#define SS    1024
#define STILE 32
#define ROW_F 148              /* 144 floats of weight + 4 pad floats (bank de-conflict) */
#define NTH   128
#define CPT   9                /* (STILE*36 chunks) / NTH threads */

/* ---------------- CDNA5 async global->LDS path (ASYNCcnt) ---------------- */

typedef int v4i __attribute__((vector_size(16)));
typedef __attribute__((address_space(1))) v4i* g_v4i_p;
typedef __attribute__((address_space(3))) v4i* l_v4i_p;

__device__ __forceinline__ void async_copy_b128(const void* gsrc, void* ldst) {
#if defined(__HIP_DEVICE_COMPILE__)
#if __has_builtin(__builtin_amdgcn_global_load_async_to_lds_b128)
  __builtin_amdgcn_global_load_async_to_lds_b128(
      (g_v4i_p)(__attribute__((address_space(1))) void*)gsrc,
      (l_v4i_p)(__attribute__((address_space(3))) void*)ldst,
      0, 0);
#else
  asm volatile("global_load_async_to_lds_b128 %0, %1, off"
               :
               : "v"((unsigned)(uintptr_t)ldst),
                 "v"((unsigned long long)(uintptr_t)gsrc)
               : "memory");
#endif
#endif
}

__device__ __forceinline__ void wait_async_le9() {
#if defined(__HIP_DEVICE_COMPILE__)
#if __has_builtin(__builtin_amdgcn_s_wait_asynccnt)
  __builtin_amdgcn_s_wait_asynccnt(CPT);
#else
  asm volatile("s_wait_asynccnt 9" ::: "memory");
#endif
#endif
}

__device__ __forceinline__ void wait_async_le0() {
#if defined(__HIP_DEVICE_COMPILE__)
#if __has_builtin(__builtin_amdgcn_s_wait_asynccnt)
  __builtin_amdgcn_s_wait_asynccnt(0);
#else
  asm volatile("s_wait_asynccnt 0" ::: "memory");
#endif
#endif
}

/* Stage one u's weight tile (32 s-rows x 144 floats, contiguous in memory)
   into LDS with 148-float padded rows.  Division-free mapping:
   thread t copies the 9 contiguous 16B chunks of c-block (t&3) of s-row (t>>2),
   i.e. 144 contiguous bytes per thread; the union is the whole 18 KB tile, so
   every 128B line is streamed exactly once. 9 async ops/wave/stage. */
__device__ __forceinline__ void issue_stage(const float* W, int v, int s0, int u,
                                            float* lbuf, int tid)
{
  const int row = tid >> 2;                 /* 0..31       */
  const int q0  = (tid & 3) * 9;            /* 0,9,18,27   */
  const char* g  = (const char*)(W + (((size_t)u * NN + v) * SS + s0 + row) * 144)
                 + q0 * 16;
  char*       lb = (char*)lbuf + row * (ROW_F * 4) + q0 * 16;
#pragma unroll
  for (int i = 0; i < CPT; ++i) {
    async_copy_b128(g, lb);
    g  += 16;
    lb += 16;
  }
}

/* -------- kernel: streaming einsum (the 151 MB weight pass) --------
   mapped[v,s,o,k] = sum_{u,c} xu[u,s,c,k] * W[u,v,s,c,o,k] * mask[u,v]*chan_map[u,v,c,o] */
extern "C" __global__ void __launch_bounds__(NTH)
pe_main(const float* __restrict__ xu, const float* __restrict__ W,
        const float* __restrict__ chan_map, const float* __restrict__ mask,
        float* __restrict__ mapped)
{
  __shared__ __align__(16) float wbuf[2][STILE * ROW_F];   /* 2 x 18.5 KB */
  __shared__ float g_lds[NN * CC * CC];                    /* mask*chan_map */

  const int tid = threadIdx.x;
  const int v   = blockIdx.y;
  const int s0  = blockIdx.x * STILE;
  const int sl  = tid & (STILE - 1);   /* s within tile */
  const int c   = tid >> 5;            /* input channel owned by this thread */

  /* kick off the first weight-tile DMA before doing anything else */
  issue_stage(W, v, s0, 0, &wbuf[0][0], tid);

  for (int idx = tid; idx < NN * CC * CC; idx += NTH) {
    const int u  = idx >> 4;
    const int ci = (idx >> 2) & 3;
    const int o  = idx & 3;
    g_lds[idx] = mask[u * NN + v] * chan_map[((u * NN + v) * CC + ci) * CC + o];
  }

  float acc[36];
#pragma unroll
  for (int i = 0; i < 36; ++i) acc[i] = 0.0f;

  for (int u = 0; u < NN; ++u) {
    const int cur = u & 1;
    if (u + 1 < NN) {
      __syncthreads();                               /* buf[cur^1] free to overwrite */
      issue_stage(W, v, s0, u + 1, &wbuf[cur ^ 1][0], tid);
      wait_async_le9();                              /* stage u's 9 copies done (in-order) */
    } else {
      wait_async_le0();
    }
    __syncthreads();                                 /* all waves' stage-u data in LDS */

    /* x values for (u, s, c, k=0..8) from L2-resident xu */
    const float4* xp = (const float4*)(xu + ((size_t)(u * SS + s0 + sl) * CC + c) * 12);
    const float4 a0 = xp[0];
    const float4 a1 = xp[1];
    const float4 a2 = xp[2];
    const float xk[9] = {a0.x, a0.y, a0.z, a0.w, a1.x, a1.y, a1.z, a1.w, a2.x};

    const float* wrow = &wbuf[cur][sl * ROW_F + c * 36];
    float4 wv[9];
#pragma unroll
    for (int q = 0; q < 9; ++q) wv[q] = ((const float4*)wrow)[q];

#pragma unroll
    for (int o = 0; o < 4; ++o) {
      const float m = g_lds[u * 16 + c * 4 + o];
#pragma unroll
      for (int k = 0; k < 9; ++k) {
        const int e = o * 9 + k;
        const float w =
            ((e & 3) == 0) ? wv[e >> 2].x :
            ((e & 3) == 1) ? wv[e >> 2].y :
            ((e & 3) == 2) ? wv[e >> 2].z : wv[e >> 2].w;
        acc[e] = fmaf(xk[k] * m, w, acc[e]);
      }
    }
  }

  /* reduce the 4 c-partials per s through LDS (reuse wbuf[0]: 4608 <= 4736 floats) */
  float* red = &wbuf[0][0];
#pragma unroll
  for (int i = 0; i < 36; ++i) red[(c * STILE + sl) * 36 + i] = acc[i];
  __syncthreads();
  if (tid < STILE) {
    float* out = mapped + ((size_t)v * SS + s0 + tid) * 36;
#pragma unroll
    for (int i = 0; i < 36; ++i) {
      out[i] = red[(0 * STILE + tid) * 36 + i] + red[(1 * STILE + tid) * 36 + i]
             + red[(2 * STILE + tid) * 36 + i] + red[(3 * STILE + tid) * 36 + i];
    }
  }
}

/* -------- kernel: unfold x -> xu[u][s][c][12] (k padded 9->12) -------- */
extern "C" __global__ void pe_unfold(const float* __restrict__ x,
                                     float* __restrict__ xu)
{
  int t = blockIdx.x * blockDim.x + threadIdx.x;
  if (t >= NN * SS * CC) return;
  int u = t >> 12;
  int s = (t >> 2) & (SS - 1);
  int c = t & 3;
  int h = s >> 5, w = s & 31;
  float* out = xu + (size_t)t * 12;
#pragma unroll
  for (int kk = 0; kk < 9; ++kk) {
    int i = kk / 3, j = kk % 3;
    int hh = h + i - 1, ww = w + j - 1;
    float val = 0.0f;
    if ((unsigned)hh < 32u && (unsigned)ww < 32u)
      val = x[((u * CC + c) * 32 + hh) * 32 + ww];
    out[kk] = val;
  }
  out[9] = 0.0f; out[10] = 0.0f; out[11] = 0.0f;
}

/* -------- kernel: fold (torch Fold reinterprets channel as (C,kh,kw)) -------- */
extern "C" __global__ void pe_fold(const float* __restrict__ mapped,
                                   float* __restrict__ out)
{
  int t = blockIdx.x * blockDim.x + threadIdx.x;
  if (t >= NN * CC * SS) return;
  int v  = t >> 12;
  int cf = (t >> 10) & 3;
  int h  = (t >> 5) & 31;
  int w  = t & 31;
  float sum = 0.0f;
#pragma unroll
  for (int i = 0; i < 3; ++i) {
    int hh = h + 1 - i;
    if ((unsigned)hh >= 32u) continue;
#pragma unroll
    for (int j = 0; j < 3; ++j) {
      int ww = w + 1 - j;
      if ((unsigned)ww >= 32u) continue;
      int ch = cf * 9 + i * 3 + j;     /* fold channel (cf,i,j)      */
      int k  = ch >> 2;                /* ufld channel was k*4 + o   */
      int o  = ch & 3;
      sum += mapped[((size_t)v * SS + hh * 32 + ww) * 36 + o * 9 + k];
    }
  }
  out[t] = sum;
}

/* ------------------------------ launcher ------------------------------ */
extern "C" void kernel_launch(void* const* d_in, const int* in_sizes, int n_in,
                              void* d_out, int out_size, void* d_ws, size_t ws_size,
                              hipStream_t stream)
{
  const float* x        = (const float*)d_in[0];   /* (16,4,32,32)            */
  const float* weight   = (const float*)d_in[1];   /* (16,16,1024,4,4,9)      */
  const float* chan_map = (const float*)d_in[2];   /* (16,16,4,4)             */
  const float* mask     = (const float*)d_in[3];   /* (16,16)                 */
  float* out = (float*)d_out;

  float* xu     = (float*)d_ws;                          /* 3,145,728 B */
  float* mapped = (float*)((char*)d_ws + 3145728);       /* 2,359,296 B */

  pe_unfold<<<256, 256, 0, stream>>>(x, xu);

  dim3 grid(SS / STILE, NN);
  pe_main<<<grid, NTH, 0, stream>>>(xu, weight, chan_map, mask, mapped);

  pe_fold<<<256, 256, 0, stream>>>(mapped, out);

  (void)in_sizes; (void)n_in; (void)out_size; (void)ws_size;
}